// GCN_GAT_33629593928257
// MI455X (gfx1250) — compile-verified
//
#include <hip/hip_runtime.h>
#include <math.h>

#define NNODES 50000
#define NEDGES 500000
#define NBATCH 256
#define FDIM   320
#define NHEADS 10
#define ENTOT  (NEDGES + NNODES)   // edges + self loops

typedef __attribute__((ext_vector_type(2))) float v2f;
typedef __attribute__((ext_vector_type(8))) float v8f;

// explicit global-address-space pointers: force global_load_* (not flat_load_*)
typedef const float __attribute__((address_space(1))) * gcfp;
typedef const v2f  __attribute__((address_space(1))) * gv2fp;

// ---------------- atomic helpers ----------------
__device__ __forceinline__ void atomAddF(float* p, float v) {
  __hip_atomic_fetch_add(p, v, __ATOMIC_RELAXED, __HIP_MEMORY_SCOPE_AGENT);
}
// float max via integer-ordered bit tricks (correct for mixed signs, -inf init)
__device__ __forceinline__ void atomMaxF(float* p, float v) {
  if (v >= 0.0f) atomicMax((int*)p, __float_as_int(v));
  else           atomicMin((unsigned int*)p, __float_as_uint(v));
}

// ---------------- utility kernels ----------------
__global__ void fill_f32(float* p, float v, int n) {
  int i = blockIdx.x * blockDim.x + threadIdx.x;
  if (i < n) p[i] = v;
}
// WT[n*K + k] = W[k*Nn + n]   (W is [K,Nn] row-major -> WT is [Nn,K] row-major)
__global__ void transpose_f32(const float* __restrict__ W, float* __restrict__ WT,
                              int K, int Nn) {
  int i = blockIdx.x * blockDim.x + threadIdx.x;
  if (i >= K * Nn) return;
  int k = i / Nn, n = i - k * Nn;
  WT[(size_t)n * K + k] = W[i];
}

// ---------------- WMMA f32 GEMM: C[M,Nn] = A[M,K] @ W[K,Nn] (+bias, opt ReLU) ----
// W is passed PRE-TRANSPOSED as WT[Nn,K], so both A and B fragments are contiguous
// 8-byte vector loads. One wave computes a 16x64 C strip (4 accumulators sharing
// one A fragment). Out-of-range rows/cols are handled by ADDRESS CLAMPING, not
// predication: garbage in A lane r only pollutes C row r; in B lane r only C
// column r — both masked at the store. EXEC stays all-ones; inner loop is
// branch-free. Requires K % 4 == 0 (call sites: 320, 640, 1500, 256, 1024).
__global__ void __launch_bounds__(128)
gemm_wmma_f32(const float* __restrict__ A, const float* __restrict__ WT,
              const float* __restrict__ bias, float* __restrict__ C,
              int M, int K, int Nn, int relu)
{
  const int lane = threadIdx.x & 31;
  const int wave = threadIdx.x >> 5;
  const int half = lane >> 4;       // K half (0: K=0,1 ; 1: K=2,3)
  const int r    = lane & 15;       // row (A) / col (B,C)
  const int tm   = blockIdx.x * 16;
  const int tn0  = (blockIdx.y * 16 + wave * 4) * 16;  // first of 4 n-tiles
  if (tn0 >= Nn) return;            // uniform per wave

  const int arow = tm + r;
  const int rowc = arow < M ? arow : (M - 1);
  const int c0 = tn0 + 0 * 16 + r, cc0 = c0 < Nn ? c0 : (Nn - 1);
  const int c1 = tn0 + 1 * 16 + r, cc1 = c1 < Nn ? c1 : (Nn - 1);
  const int c2 = tn0 + 2 * 16 + r, cc2 = c2 < Nn ? c2 : (Nn - 1);
  const int c3 = tn0 + 3 * 16 + r, cc3 = c3 < Nn ? c3 : (Nn - 1);

  gcfp ap  = (gcfp)(A  + (size_t)rowc * K + 2 * half);
  gcfp bp0 = (gcfp)(WT + (size_t)cc0  * K + 2 * half);
  gcfp bp1 = (gcfp)(WT + (size_t)cc1  * K + 2 * half);
  gcfp bp2 = (gcfp)(WT + (size_t)cc2  * K + 2 * half);
  gcfp bp3 = (gcfp)(WT + (size_t)cc3  * K + 2 * half);

  v8f acc0 = {}, acc1 = {}, acc2 = {}, acc3 = {};
  for (int k0 = 0; k0 < K; k0 += 4) {
    v2f av = *(gv2fp)ap;
    v2f b0 = *(gv2fp)bp0;
    v2f b1 = *(gv2fp)bp1;
    v2f b2 = *(gv2fp)bp2;
    v2f b3 = *(gv2fp)bp3;
    acc0 = __builtin_amdgcn_wmma_f32_16x16x4_f32(false, av, false, b0, (short)0, acc0, false, false);
    acc1 = __builtin_amdgcn_wmma_f32_16x16x4_f32(false, av, false, b1, (short)0, acc1, false, false);
    acc2 = __builtin_amdgcn_wmma_f32_16x16x4_f32(false, av, false, b2, (short)0, acc2, false, false);
    acc3 = __builtin_amdgcn_wmma_f32_16x16x4_f32(false, av, false, b3, (short)0, acc3, false, false);
    ap += 4; bp0 += 4; bp1 += 4; bp2 += 4; bp3 += 4;
  }

  v8f accs[4] = {acc0, acc1, acc2, acc3};
  const int cols[4] = {c0, c1, c2, c3};
#pragma unroll
  for (int j = 0; j < 4; ++j) {
    int col = cols[j];
    if (col < Nn) {
      float bb = bias ? bias[col] : 0.0f;
#pragma unroll
      for (int i = 0; i < 8; ++i) {
        int orow = tm + 8 * half + i;         // C layout: VGPR i -> M = i + 8*half
        if (orow < M) {
          float v = accs[j][i] + bb;
          if (relu) v = fmaxf(v, 0.0f);
          C[(size_t)orow * Nn + col] = v;
        }
      }
    }
  }
}

// ---------------- GCN ----------------
__global__ void gcn_deg(const int* __restrict__ ei, float* deg) {
  int e = blockIdx.x * blockDim.x + threadIdx.x;
  if (e >= NEDGES) return;
  atomAddF(&deg[ei[NEDGES + e]], 1.0f);
}
__global__ void inv_sqrt(float* d, int n) {
  int i = blockIdx.x * blockDim.x + threadIdx.x;
  if (i < n) d[i] = rsqrtf(d[i]);
}
__global__ void gcn_self(const float* __restrict__ H1, const float* __restrict__ dinv,
                         float* __restrict__ X1) {
  int i = blockIdx.x * blockDim.x + threadIdx.x;   // n*F+f
  if (i >= NNODES * FDIM) return;
  int n = i / FDIM;
  float di = dinv[n];
  X1[i] = H1[i] * di * di;                         // self-loop term
}
__global__ void gcn_edge(const float* __restrict__ H1, const float* __restrict__ dinv,
                         const int* __restrict__ ei, float* __restrict__ X1) {
  int e = blockIdx.x, f = threadIdx.x;             // block = 320 threads
  int s = ei[e], d = ei[NEDGES + e];
  float coef = dinv[s] * dinv[d];
  atomAddF(&X1[(size_t)d * FDIM + f], H1[(size_t)s * FDIM + f] * coef);
}
__global__ void bias_act(float* X, const float* __restrict__ b, int total, int cols, int relu) {
  int i = blockIdx.x * blockDim.x + threadIdx.x;
  if (i >= total) return;
  float v = X[i] + b[i % cols];
  if (relu) v = fmaxf(v, 0.0f);
  X[i] = v;
}

// ---------------- GAT ----------------
__global__ void gat_coef(const float* __restrict__ H2, const float* __restrict__ as_w,
                         const float* __restrict__ ad_w, float* a_s, float* a_d) {
  int i = blockIdx.x * blockDim.x + threadIdx.x;   // n*H + h
  if (i >= NNODES * NHEADS) return;
  int n = i / NHEADS, h = i - n * NHEADS;
  const float* hp = H2 + (size_t)n * FDIM + h * 32;
  float s = 0.0f, d = 0.0f;
#pragma unroll
  for (int c = 0; c < 32; ++c) { float v = hp[c]; s += v * as_w[h * 32 + c]; d += v * ad_w[h * 32 + c]; }
  a_s[i] = s; a_d[i] = d;
}
__device__ __forceinline__ void edgeSD(const int* ei, int e, int& s, int& d) {
  if (e < NEDGES) { s = ei[e]; d = ei[NEDGES + e]; } else { s = e - NEDGES; d = s; }
}
__global__ void gat_edge_max(const int* __restrict__ ei, const float* __restrict__ a_s,
                             const float* __restrict__ a_d, float* __restrict__ ebuf,
                             float* __restrict__ m) {
  int i = blockIdx.x * blockDim.x + threadIdx.x;   // e*H + h
  if (i >= ENTOT * NHEADS) return;
  int e = i / NHEADS, h = i - e * NHEADS;
  int s, d; edgeSD(ei, e, s, d);
  float v = a_s[s * NHEADS + h] + a_d[d * NHEADS + h];
  v = v > 0.0f ? v : 0.2f * v;                     // leaky relu, slope 0.2
  ebuf[i] = v;
  atomMaxF(&m[d * NHEADS + h], v);
}
__global__ void gat_edge_exp(const int* __restrict__ ei, float* __restrict__ ebuf,
                             const float* __restrict__ m, float* __restrict__ denom) {
  int i = blockIdx.x * blockDim.x + threadIdx.x;
  if (i >= ENTOT * NHEADS) return;
  int e = i / NHEADS, h = i - e * NHEADS;
  int s, d; edgeSD(ei, e, s, d);
  float p = expf(ebuf[i] - m[d * NHEADS + h]);
  ebuf[i] = p;
  atomAddF(&denom[d * NHEADS + h], p);
}
__global__ void gat_edge_agg(const int* __restrict__ ei, const float* __restrict__ H2,
                             const float* __restrict__ ebuf, const float* __restrict__ denom,
                             float* __restrict__ X2) {
  int e = blockIdx.x, f = threadIdx.x, h = f >> 5; // block = 320 threads, head = f/32
  int s, d; edgeSD(ei, e, s, d);
  float alpha = ebuf[e * NHEADS + h] / denom[d * NHEADS + h];
  atomAddF(&X2[(size_t)d * FDIM + f], H2[(size_t)s * FDIM + f] * alpha);
}

// ---------------- pooling / glue ----------------
__global__ void pool_acc(const float* __restrict__ X2, const int* __restrict__ batch,
                         float* gmp, float* gap, float* cnt) {
  int n = blockIdx.x, f = threadIdx.x;
  int b = batch[n];
  float v = X2[(size_t)n * FDIM + f];
  atomMaxF(&gmp[b * FDIM + f], v);
  atomAddF(&gap[b * FDIM + f], v);
  if (f == 0) atomAddF(&cnt[b], 1.0f);
}
__global__ void pool_fin(const float* gmp, const float* gap, const float* cnt, float* g) {
  int b = blockIdx.x, f = threadIdx.x;
  float c = fmaxf(cnt[b], 1.0f);
  g[b * 2 * FDIM + f]        = gmp[b * FDIM + f];
  g[b * 2 * FDIM + FDIM + f] = gap[b * FDIM + f] / c;
}
__global__ void concat_g(const float* ga, const float* gb, float* xc) {
  int b = blockIdx.x, c = threadIdx.x;             // block = 256
  xc[b * 256 + c] = (c < 128) ? ga[b * 128 + c] : gb[b * 128 + (c - 128)];
}
__global__ void out_head(const float* __restrict__ Y, const float* __restrict__ Wo,
                         const float* __restrict__ bo, float* __restrict__ out) {
  int idx = blockIdx.x * blockDim.x + threadIdx.x; // 512 threads
  if (idx >= NBATCH * 2) return;
  int row = idx >> 1, c = idx & 1;
  float acc = bo[c];
  const float* yp = Y + (size_t)row * 512;
  for (int k = 0; k < 512; ++k) acc += yp[k] * Wo[k * 2 + c];
  out[row * 2 + c] = acc;
}

// ---------------- host side ----------------
static inline void fillh(float* p, float v, int n, hipStream_t s) {
  fill_f32<<<(n + 255) / 256, 256, 0, s>>>(p, v, n);
}
static inline void transposeh(const float* W, float* WT, int K, int Nn, hipStream_t s) {
  int total = K * Nn;
  transpose_f32<<<(total + 255) / 256, 256, 0, s>>>(W, WT, K, Nn);
}
// WT must be the [Nn,K] transposed weight
static inline void launch_gemm(const float* A, const float* WT, const float* bias, float* C,
                               int M, int K, int Nn, int relu, hipStream_t s) {
  int ntiles = (Nn + 15) / 16;
  dim3 grid((M + 15) / 16, (ntiles + 15) / 16);    // block: 4 waves x (4 tiles each)
  gemm_wmma_f32<<<grid, 128, 0, s>>>(A, WT, bias, C, M, K, Nn, relu);
}

struct Ws {
  float *buf0, *buf1, *ebuf, *dinv, *a_s, *a_d, *mmax, *denom;
  float *gmp, *gap, *cnt, *g, *g1;
  float *wgcnT, *wgatT, *wfcg1T, *wfcg2T, *wfc1T, *wfc2T;
};

static void run_branch(const float* x, const int* ei, const int* batch,
                       const float* b_gcn,
                       const float* att_src, const float* att_dst, const float* b_gat,
                       const float* b_fc_g1, const float* b_fc_g2,
                       float* g2_out, const Ws& w, hipStream_t s)
{
  // GCN: H1 = x @ W_gcn  (buf0)
  launch_gemm(x, w.wgcnT, nullptr, w.buf0, NNODES, FDIM, FDIM, 0, s);
  // degrees (self loop => init 1), dinv = rsqrt(deg)
  fillh(w.dinv, 1.0f, NNODES, s);
  gcn_deg<<<(NEDGES + 255) / 256, 256, 0, s>>>(ei, w.dinv);
  inv_sqrt<<<(NNODES + 255) / 256, 256, 0, s>>>(w.dinv, NNODES);
  // aggregate into buf1, then bias+relu -> X1
  gcn_self<<<(NNODES * FDIM + 255) / 256, 256, 0, s>>>(w.buf0, w.dinv, w.buf1);
  gcn_edge<<<NEDGES, FDIM, 0, s>>>(w.buf0, w.dinv, ei, w.buf1);
  bias_act<<<(NNODES * FDIM + 255) / 256, 256, 0, s>>>(w.buf1, b_gcn, NNODES * FDIM, FDIM, 1);
  // GAT: H2 = X1 @ W_gat (reuse buf0)
  launch_gemm(w.buf1, w.wgatT, nullptr, w.buf0, NNODES, FDIM, FDIM, 0, s);
  gat_coef<<<(NNODES * NHEADS + 255) / 256, 256, 0, s>>>(w.buf0, att_src, att_dst, w.a_s, w.a_d);
  fillh(w.mmax, -INFINITY, NNODES * NHEADS, s);
  fillh(w.denom, 0.0f, NNODES * NHEADS, s);
  gat_edge_max<<<(ENTOT * NHEADS + 255) / 256, 256, 0, s>>>(ei, w.a_s, w.a_d, w.ebuf, w.mmax);
  gat_edge_exp<<<(ENTOT * NHEADS + 255) / 256, 256, 0, s>>>(ei, w.ebuf, w.mmax, w.denom);
  fillh(w.buf1, 0.0f, NNODES * FDIM, s);
  gat_edge_agg<<<ENTOT, FDIM, 0, s>>>(ei, w.buf0, w.ebuf, w.denom, w.buf1);
  bias_act<<<(NNODES * FDIM + 255) / 256, 256, 0, s>>>(w.buf1, b_gat, NNODES * FDIM, FDIM, 1);
  // pooling -> g [B, 640]
  fillh(w.gmp, -INFINITY, NBATCH * FDIM, s);
  fillh(w.gap, 0.0f, NBATCH * FDIM, s);
  fillh(w.cnt, 0.0f, NBATCH, s);
  pool_acc<<<NNODES, FDIM, 0, s>>>(w.buf1, batch, w.gmp, w.gap, w.cnt);
  pool_fin<<<NBATCH, FDIM, 0, s>>>(w.gmp, w.gap, w.cnt, w.g);
  // branch FCs
  launch_gemm(w.g,  w.wfcg1T, b_fc_g1, w.g1,   NBATCH, 2 * FDIM, 1500, 1, s);
  launch_gemm(w.g1, w.wfcg2T, b_fc_g2, g2_out, NBATCH, 1500,     128,  0, s);
}

extern "C" void kernel_launch(void* const* d_in, const int* in_sizes, int n_in,
                              void* d_out, int out_size, void* d_ws, size_t ws_size,
                              hipStream_t stream) {
  (void)in_sizes; (void)n_in; (void)out_size; (void)ws_size;
  const float* x_a     = (const float*)d_in[0];
  const float* x_b     = (const float*)d_in[1];
  const int*   ei_a    = (const int*)d_in[2];
  const int*   ei_b    = (const int*)d_in[3];
  const int*   batch_a = (const int*)d_in[4];
  const int*   batch_b = (const int*)d_in[5];
  const float* W_gcn   = (const float*)d_in[6];
  const float* b_gcn   = (const float*)d_in[7];
  const float* W_gat   = (const float*)d_in[8];
  const float* att_src = (const float*)d_in[9];
  const float* att_dst = (const float*)d_in[10];
  const float* b_gat   = (const float*)d_in[11];
  const float* W_fc_g1 = (const float*)d_in[12];
  const float* b_fc_g1 = (const float*)d_in[13];
  const float* W_fc_g2 = (const float*)d_in[14];
  const float* b_fc_g2 = (const float*)d_in[15];
  const float* W_fc1   = (const float*)d_in[16];
  const float* b_fc1   = (const float*)d_in[17];
  const float* W_fc2   = (const float*)d_in[18];
  const float* b_fc2   = (const float*)d_in[19];
  const float* W_out   = (const float*)d_in[20];
  const float* b_out   = (const float*)d_in[21];

  char* p = (char*)d_ws;
  auto alloc = [&](size_t elems) -> float* {
    float* r = (float*)p;
    p += (elems * sizeof(float) + 255) & ~(size_t)255;
    return r;
  };
  Ws w;
  w.buf0   = alloc((size_t)NNODES * FDIM);    // 64 MB (H1 / H2)
  w.buf1   = alloc((size_t)NNODES * FDIM);    // 64 MB (X1 / X2)
  w.ebuf   = alloc((size_t)ENTOT * NHEADS);   // 22 MB (edge logits / probs)
  w.dinv   = alloc(NNODES);
  w.a_s    = alloc(NNODES * NHEADS);
  w.a_d    = alloc(NNODES * NHEADS);
  w.mmax   = alloc(NNODES * NHEADS);
  w.denom  = alloc(NNODES * NHEADS);
  w.gmp    = alloc(NBATCH * FDIM);
  w.gap    = alloc(NBATCH * FDIM);
  w.cnt    = alloc(NBATCH);
  w.g      = alloc(NBATCH * 2 * FDIM);
  w.g1     = alloc(NBATCH * 1500);
  w.wgcnT  = alloc((size_t)FDIM * FDIM);
  w.wgatT  = alloc((size_t)FDIM * FDIM);
  w.wfcg1T = alloc((size_t)2 * FDIM * 1500);
  w.wfcg2T = alloc((size_t)1500 * 128);
  w.wfc1T  = alloc((size_t)256 * 1024);
  w.wfc2T  = alloc((size_t)1024 * 512);
  float* g2a = alloc(NBATCH * 128);
  float* g2b = alloc(NBATCH * 128);
  float* xc  = alloc(NBATCH * 256);
  float* y1  = alloc(NBATCH * 1024);
  float* y2  = alloc(NBATCH * 512);

  // one-time (per call) weight transposes; weights shared by both branches
  transposeh(W_gcn,   w.wgcnT,  FDIM,     FDIM, stream);
  transposeh(W_gat,   w.wgatT,  FDIM,     FDIM, stream);
  transposeh(W_fc_g1, w.wfcg1T, 2 * FDIM, 1500, stream);
  transposeh(W_fc_g2, w.wfcg2T, 1500,     128,  stream);
  transposeh(W_fc1,   w.wfc1T,  256,      1024, stream);
  transposeh(W_fc2,   w.wfc2T,  1024,     512,  stream);

  run_branch(x_a, ei_a, batch_a, b_gcn, att_src, att_dst, b_gat,
             b_fc_g1, b_fc_g2, g2a, w, stream);
  run_branch(x_b, ei_b, batch_b, b_gcn, att_src, att_dst, b_gat,
             b_fc_g1, b_fc_g2, g2b, w, stream);

  concat_g<<<NBATCH, 256, 0, stream>>>(g2a, g2b, xc);
  launch_gemm(xc, w.wfc1T, b_fc1, y1, NBATCH, 256,  1024, 1, stream);
  launch_gemm(y1, w.wfc2T, b_fc2, y2, NBATCH, 1024, 512,  1, stream);
  out_head<<<2, 256, 0, stream>>>(y2, W_out, b_out, (float*)d_out);
}